// TwoDQClassifier_37220186587899
// MI455X (gfx1250) — compile-verified
//
#include <hip/hip_runtime.h>

// TwoDQClassifier on gfx1250: state-vector simulation of the 12-qubit circuit.
// out = U @ e0 is just the state propagated through 4 layers, so we never build
// the 4096x4096 unitary. Each layer's 12 commuting RY gates form A (x) B (x) C
// (three 16x16 Kronecker factors); applying them = 3 mode contractions of the
// 16x16x16 state tensor, each done with chained V_WMMA_F32_16X16X4_F32 (exact
// fp32). The CNOT ring is a bit-permutation gather. Everything lives in LDS of
// a single workgroup (8 waves) -- the problem is ~1.6 MFLOP / 33 KB: pure
// latency, so one launch + zero global round-trips is optimal on MI455X.

#define NQ     12
#define DIM    4096
#define DEPTH  4
#define ALPHA  0.1f

typedef __attribute__((ext_vector_type(2))) float v2f;
typedef __attribute__((ext_vector_type(8))) float v8f;

__global__ __launch_bounds__(256) void tdq_classifier_kernel(
    const float* __restrict__ x,      // [12]
    const float* __restrict__ theta,  // [4][12]
    const float* __restrict__ w_out,  // [2][4096]
    const float* __restrict__ b_out,  // [2]
    float* __restrict__ out)          // [2 + 4096]
{
    __shared__ float S0[DIM];           // 16 KB
    __shared__ float S1[DIM];           // 16 KB
    __shared__ float Ga[256], Gb[256], Gc[256];   // 16x16 Kronecker factors
    __shared__ float CS[DEPTH * NQ], SN[DEPTH * NQ];
    __shared__ float red[512];

    const int tid  = threadIdx.x;
    const int lane = tid & 31;
    const int wave = tid >> 5;        // 0..7
    const int l    = lane & 15;       // 0..15
    const int half = lane >> 4;       // 0 or 1

    // ---- half-angle cos/sin for all 48 angles: a = theta + ALPHA*x --------
    if (tid < DEPTH * NQ) {
        int q = tid % NQ;
        float ang = 0.5f * (theta[tid] + ALPHA * x[q]);
        CS[tid] = cosf(ang);
        SN[tid] = sinf(ang);
    }
    // ---- state = |0...0> --------------------------------------------------
    for (int i = tid; i < DIM; i += 256) S0[i] = (i == 0) ? 1.0f : 0.0f;
    __syncthreads();

    for (int layer = 0; layer < DEPTH; ++layer) {
        // ---- build the three 16x16 Kronecker factors ----------------------
        // group g covers qubits 4g..4g+3; qubit 4g+j is bit (3-j) of the
        // group index.  RY entry: (rb==cb)?cos : (rb? +sin : -sin).
        {
            const int r = tid >> 4;     // 0..15
            const int c = tid & 15;     // 0..15
            float pa = 1.0f, pb = 1.0f, pc = 1.0f;
            #pragma unroll
            for (int j = 0; j < 4; ++j) {
                int rb = (r >> (3 - j)) & 1;
                int cb = (c >> (3 - j)) & 1;
                int q0 = layer * NQ + j;
                float ea = (rb == cb) ? CS[q0]      : (rb ? SN[q0]      : -SN[q0]);
                float eb = (rb == cb) ? CS[q0 + 4]  : (rb ? SN[q0 + 4]  : -SN[q0 + 4]);
                float ec = (rb == cb) ? CS[q0 + 8]  : (rb ? SN[q0 + 8]  : -SN[q0 + 8]);
                pa *= ea; pb *= eb; pc *= ec;
            }
            Ga[tid] = pa; Gb[tid] = pb; Gc[tid] = pc;
        }
        __syncthreads();

        // State tensor S[a][b][c], linear index i = a*256 + b*16 + c.
        // Fragment mapping for V_WMMA_F32_16X16X4_F32 (per ISA 7.12.2):
        //   A 16x4 : lane l+16h, vgpr j -> A[l][2h+j]
        //   B 4x16 : lane l+16h, vgpr j -> B[2h+j][l]
        //   C/D    : lane l+16h, vgpr j -> D[8h+j][l]

        // ---- contraction 1 (mode a): T1[a'][b][c] = sum_a A[a'][a] S[abc] --
        // 16 column tiles (n = b*16+c); wave w owns tiles 2w, 2w+1.
        for (int tt = 0; tt < 2; ++tt) {
            const int t = wave * 2 + tt;
            v8f acc = {};
            #pragma unroll
            for (int p = 0; p < 4; ++p) {
                const int k0 = p * 4 + half * 2;
                v2f a, b;
                a[0] = Ga[l * 16 + k0];
                a[1] = Ga[l * 16 + k0 + 1];
                b[0] = S0[(k0    ) * 256 + t * 16 + l];
                b[1] = S0[(k0 + 1) * 256 + t * 16 + l];
                acc = __builtin_amdgcn_wmma_f32_16x16x4_f32(
                          false, a, false, b, (short)0, acc, false, false);
            }
            #pragma unroll
            for (int j = 0; j < 8; ++j)
                S1[(half * 8 + j) * 256 + t * 16 + l] = acc[j];
        }
        __syncthreads();

        // ---- contraction 2 (mode b): T2[a'][b'][c] = sum_b B[b'][b] T1 ----
        // 16 independent 16x16 matmuls, tile index = a'.
        for (int tt = 0; tt < 2; ++tt) {
            const int t = wave * 2 + tt;
            v8f acc = {};
            #pragma unroll
            for (int p = 0; p < 4; ++p) {
                const int k0 = p * 4 + half * 2;
                v2f a, b;
                a[0] = Gb[l * 16 + k0];
                a[1] = Gb[l * 16 + k0 + 1];
                b[0] = S1[t * 256 + (k0    ) * 16 + l];
                b[1] = S1[t * 256 + (k0 + 1) * 16 + l];
                acc = __builtin_amdgcn_wmma_f32_16x16x4_f32(
                          false, a, false, b, (short)0, acc, false, false);
            }
            #pragma unroll
            for (int j = 0; j < 8; ++j)
                S0[t * 256 + (half * 8 + j) * 16 + l] = acc[j];
        }
        __syncthreads();

        // ---- contraction 3 (mode c): S'[m][c'] = (rows m) @ C^T -----------
        // m = a'*16 + b' (256 rows, 16 row tiles); A-operand = state rows.
        for (int tt = 0; tt < 2; ++tt) {
            const int t = wave * 2 + tt;
            v8f acc = {};
            #pragma unroll
            for (int p = 0; p < 4; ++p) {
                const int k0 = p * 4 + half * 2;
                v2f a, b;
                a[0] = S0[(t * 16 + l) * 16 + k0];
                a[1] = S0[(t * 16 + l) * 16 + k0 + 1];
                b[0] = Gc[l * 16 + k0];      // C^T[k][col] = C[col][k]
                b[1] = Gc[l * 16 + k0 + 1];
                acc = __builtin_amdgcn_wmma_f32_16x16x4_f32(
                          false, a, false, b, (short)0, acc, false, false);
            }
            #pragma unroll
            for (int j = 0; j < 8; ++j)
                S1[(t * 16 + half * 8 + j) * 16 + l] = acc[j];
        }
        __syncthreads();

        // ---- CNOT ring: new[i] = old[g1(g2(...g12(i)))] -------------------
        // gate q = CNOT(q, (q+1)%12); qubit q <-> bit (11-q).  Last-applied
        // gate (q=11) is innermost.
        for (int i = tid; i < DIM; i += 256) {
            int jdx = i;
            #pragma unroll
            for (int q = NQ - 1; q >= 0; --q) {
                int cbit = (jdx >> (11 - q)) & 1;
                jdx ^= cbit << (11 - ((q + 1) % NQ));
            }
            S0[i] = S1[jdx];
        }
        __syncthreads();
    }

    // ---- readout: logits[k] = sum_i s[i]^2 * w_out[k][i] + b_out[k] -------
    float p0 = 0.0f, p1 = 0.0f;
    for (int i = tid; i < DIM; i += 256) {
        float s = S0[i];
        out[2 + i] = s;                 // second tuple element: the state
        float pr = s * s;
        p0 += pr * w_out[i];
        p1 += pr * w_out[DIM + i];
    }
    red[tid]       = p0;
    red[256 + tid] = p1;
    __syncthreads();
    #pragma unroll
    for (int off = 128; off > 0; off >>= 1) {
        if (tid < off) {
            red[tid]       += red[tid + off];
            red[256 + tid] += red[256 + tid + off];
        }
        __syncthreads();
    }
    if (tid == 0) {
        out[0] = red[0]   + b_out[0];
        out[1] = red[256] + b_out[1];
    }
}

extern "C" void kernel_launch(void* const* d_in, const int* in_sizes, int n_in,
                              void* d_out, int out_size, void* d_ws, size_t ws_size,
                              hipStream_t stream) {
    (void)in_sizes; (void)n_in; (void)out_size; (void)d_ws; (void)ws_size;
    const float* x     = (const float*)d_in[0];   // [12]
    const float* theta = (const float*)d_in[1];   // [4*12]
    const float* w     = (const float*)d_in[2];   // [2*4096]
    const float* b     = (const float*)d_in[3];   // [2]
    float* out = (float*)d_out;                   // [2 + 4096]
    tdq_classifier_kernel<<<1, 256, 0, stream>>>(x, theta, w, b, out);
}